// attention_embedding_70540542869848
// MI455X (gfx1250) — compile-verified
//
#include <hip/hip_runtime.h>

// Clang ext-vector types (HIP's float4 is a struct; these work with
// __builtin_nontemporal_load and the WMMA builtins).
typedef float v2f __attribute__((ext_vector_type(2)));
typedef float v4f __attribute__((ext_vector_type(4)));
typedef float v8f __attribute__((ext_vector_type(8)));

#define B_ 4
#define N_ 512
#define D_ 128

// ---------------------------------------------------------------------------
// Kernel 1: out[b] = att[b] (512x512) @ v[b] (512x128), fp32, via
// V_WMMA_F32_16X16X4_F32. One wave32 per 16x16 output tile.
//   A 16x4 layout (ISA 7.12.2): lane = {m=lane&15, half=lane>>4};
//     VGPR0 = A[m][k + 2*half], VGPR1 = A[m][k + 2*half + 1]
//   B 4x16: row K striped across lanes within a VGPR (same half-split on K)
//   C/D 16x16: VGPR r = row (r + 8*half), col = lane&15
// ---------------------------------------------------------------------------
__global__ __launch_bounds__(128) void wmma_gemm_av(const float* __restrict__ att,
                                                    const float* __restrict__ v,
                                                    float* __restrict__ out) {
    const int lane = threadIdx.x & 31;
    const int gw   = blockIdx.x * 4 + (threadIdx.x >> 5);  // 1024 waves total
    const int b    = gw >> 8;                              // 256 tiles / batch
    const int rem  = gw & 255;
    const int i0   = (rem >> 3) << 4;                      // 32 i-tiles
    const int d0   = (rem & 7) << 4;                       // 8 d-tiles

    const int m    = lane & 15;
    const int n    = lane & 15;
    const int half = lane >> 4;  // 0 -> K pair {0,1}; 1 -> K pair {2,3}

    const float* rowA = att + ((size_t)b * N_ + (i0 + m)) * N_;      // A row, stride 1 in K
    const float* colB = v + ((size_t)b * N_) * D_ + d0 + n;          // B col, stride D_ in K

    v8f c = {};
#pragma unroll 4
    for (int k = 0; k < N_; k += 4) {
        // A fragment: two consecutive K values -> one 8-byte load
        v2f a = *(const v2f*)(rowA + k + 2 * half);
        // B fragment: rows k+2*half and k+2*half+1 of v[b], column d0+n
        const float* bp = colB + (size_t)(k + 2 * half) * D_;
        v2f bm;
        bm.x = bp[0];
        bm.y = bp[D_];
        // (neg_a, A, neg_b, B, c_mod, C, reuse_a, reuse_b)
        c = __builtin_amdgcn_wmma_f32_16x16x4_f32(false, a, false, bm,
                                                  (short)0, c, false, false);
    }

    float* orow = out + ((size_t)b * N_ + i0 + 8 * half) * D_ + d0 + n;
#pragma unroll
    for (int r = 0; r < 8; ++r)
        orow[(size_t)r * D_] = c[r];
}

// ---------------------------------------------------------------------------
// Kernel 2: out[b,i,:] += sum_j att[b,i,j] * (rel[b,i,j,:] + edge[b,i,j,:])
// One 256-thread workgroup (8 waves) per (b,i); streams 512 KB/WG of
// single-use data with non-temporal b128 loads. att[b,i,j] is wave-uniform.
// ---------------------------------------------------------------------------
__global__ __launch_bounds__(256) void stream_weighted_sum(const float* __restrict__ att,
                                                           const float* __restrict__ rel,
                                                           const float* __restrict__ edge,
                                                           float* __restrict__ out) {
    __shared__ v4f red[8][32];

    const int t  = threadIdx.x;
    const int d4 = t & 31;   // float4 column within D (32 * 4 = 128)
    const int jg = t >> 5;   // j-group == wave id (att scalar is uniform per wave)
    const int bi = blockIdx.x;  // flattened b*N + i

    const v4f* rel4  = (const v4f*)rel;
    const v4f* edge4 = (const v4f*)edge;
    const size_t rowbase   = (size_t)bi * N_ * (D_ / 4);
    const float* attrow    = att + (size_t)bi * N_;

    v4f acc = {};
    for (int j = jg; j < N_; j += 8) {
        const float s = attrow[j];  // uniform across the wave
        const size_t idx = rowbase + (size_t)j * (D_ / 4) + d4;
        v4f r = __builtin_nontemporal_load(rel4 + idx);
        v4f e = __builtin_nontemporal_load(edge4 + idx);
        acc += s * (r + e);
    }

    red[jg][d4] = acc;
    __syncthreads();

    if (jg == 0) {
        v4f tot = acc;
#pragma unroll
        for (int g = 1; g < 8; ++g) tot += red[g][d4];
        v4f* op = (v4f*)(out + (size_t)bi * D_) + d4;
        *op = *op + tot;  // kernel 1 already wrote the att@v term
    }
}

extern "C" void kernel_launch(void* const* d_in, const int* in_sizes, int n_in,
                              void* d_out, int out_size, void* d_ws, size_t ws_size,
                              hipStream_t stream) {
    (void)in_sizes; (void)n_in; (void)out_size; (void)d_ws; (void)ws_size;

    const float* att  = (const float*)d_in[0];  // [B,N,N]
    const float* val  = (const float*)d_in[1];  // [B,N,D]
    const float* rel  = (const float*)d_in[2];  // [B,N,N,D]
    const float* edge = (const float*)d_in[3];  // [B,N,N,D]
    float* out = (float*)d_out;                 // [B,N,D]

    // 1) out = att @ v  (WMMA fp32 path)
    wmma_gemm_av<<<256, 128, 0, stream>>>(att, val, out);
    // 2) out += sum_j att * (rel + edge)  (bandwidth-bound streaming path)
    stream_weighted_sum<<<B_ * N_, 256, 0, stream>>>(att, rel, edge, out);
}